// MSD_computer_75084618269459
// MI455X (gfx1250) — compile-verified
//
#include <hip/hip_runtime.h>
#include <hip/hip_bf16.h>

typedef __attribute__((ext_vector_type(2))) float v2f;
typedef __attribute__((ext_vector_type(8))) float v8f;

#define T_FRAMES 4096
#define N_ATOMS  4096
#define ROW      (N_ATOMS * 3)      // 12288 floats per frame
#define TD_MAX_  512
#define N_ENS    64
#define N_ORIG   56                 // len(arange(0, 4096-512, 64))
#define WAVES    8
#define KSLICE   (ROW / WAVES)      // 1536 floats per wave => 384 WMMA steps

// One workgroup = (origin m, lag block L of 16 lags).
// Wave w reduces K-slice [w*1536, (w+1)*1536) of the 12288-long dot products.
// WMMA computes C = D * D^T over each K chunk (A and B are the SAME registers:
// f32 A-layout lane=M / B-layout lane=N with identical K striping), so
// C[r][r] accumulates sum_j (x[t0+16L+r][j] - x[t0][j])^2.
__global__ __launch_bounds__(256) void msd_wmma_kernel(const float* __restrict__ x,
                                                       float* __restrict__ partial) {
  __shared__ float sOrig[ROW];          // 48 KB: origin frame
  __shared__ float diagBuf[WAVES * 16];

  const int m  = blockIdx.x;            // origin index (0..55)
  const int L  = blockIdx.y;            // lag block   (0..31)
  const int t0 = m * N_ENS;

  // Stage origin frame into LDS, coalesced b128 loads (12 iters, uniform).
  const float4* __restrict__ src4 = reinterpret_cast<const float4*>(x + (size_t)t0 * ROW);
  float4* dst4 = reinterpret_cast<float4*>(sOrig);
  for (int i = threadIdx.x; i < ROW / 4; i += 256) dst4[i] = src4[i];
  __syncthreads();

  const int lane = threadIdx.x & 31;
  const int wave = threadIdx.x >> 5;
  const int r    = lane & 15;           // A-matrix row M = lag within block
  const int kk   = (lane >> 4) << 1;    // K sub-offset: lanes 0-15 -> K0,K1; 16-31 -> K2,K3
  const int kb   = wave * KSLICE + kk;  // this wave's K-slice base (+ lane sub-offset)

  // Moving frame row for this lane's matrix row: s = t0 + 16L + r
  const float* __restrict__ Ym = x + (size_t)(t0 + 16 * L + r) * ROW + kb;

  v8f acc = {0.f, 0.f, 0.f, 0.f, 0.f, 0.f, 0.f, 0.f};
  // Compile-time constant trip count (384): scalar loop, EXEC untouched.
#pragma unroll 4
  for (int i = 0; i < KSLICE; i += 4) {
    v2f a = *reinterpret_cast<const v2f*>(Ym + i);   // global_load_b64 (per-lane row)
    v2f o;
    o.x = sOrig[kb + i];                             // LDS (broadcast within half-wave)
    o.y = sOrig[kb + i + 1];
    v2f d = a - o;                                   // fuses to v_pk_add_f32 neg
    // 8 args: (neg_a, A, neg_b, B, c_mod, C, reuse_a, reuse_b); A == B -> D*D^T
    acc = __builtin_amdgcn_wmma_f32_16x16x4_f32(false, d, false, d, (short)0, acc,
                                                false, false);
  }

  // Extract diagonal of the 16x16 f32 C tile:
  //   diag r<8  -> lane r,      vgpr r
  //   diag r>=8 -> lane r+16,   vgpr r-8
  union { v8f v; float f[8]; } cc;
  cc.v = acc;
  const bool isDiag = (lane < 8) || (lane >= 24);
  const int  gsel   = (lane < 8) ? lane : (lane - 24);   // which accumulator vgpr
  const int  rsel   = (lane < 8) ? lane : (lane - 16);   // which lag within block
  float dv = cc.f[0];
#pragma unroll
  for (int g = 1; g < 8; ++g) dv = (gsel == g) ? cc.f[g] : dv;  // cndmask chain
  if (isDiag) diagBuf[wave * 16 + rsel] = dv;
  __syncthreads();

  if (threadIdx.x < 16) {
    float s = 0.f;
#pragma unroll
    for (int w = 0; w < WAVES; ++w) s += diagBuf[w * 16 + threadIdx.x];
    partial[(size_t)m * TD_MAX_ + L * 16 + threadIdx.x] = s;
  }
}

__global__ void msd_finalize_kernel(const float* __restrict__ partial,
                                    float* __restrict__ out) {
  int td = blockIdx.x * blockDim.x + threadIdx.x;
  if (td < TD_MAX_) {
    float s = 0.f;
#pragma unroll 8
    for (int mm = 0; mm < N_ORIG; ++mm) s += partial[(size_t)mm * TD_MAX_ + td];
    out[td] = s * (1.0f / ((float)N_ORIG * (float)ROW));
  }
}

extern "C" void kernel_launch(void* const* d_in, const int* in_sizes, int n_in,
                              void* d_out, int out_size, void* d_ws, size_t ws_size,
                              hipStream_t stream) {
  const float* x = (const float*)d_in[0];       // [4096, 4096, 3] f32
  float* out     = (float*)d_out;               // [512] f32
  float* partial = (float*)d_ws;                // [56, 512] f32 scratch

  dim3 grid(N_ORIG, TD_MAX_ / 16);              // 56 x 32 workgroups
  msd_wmma_kernel<<<grid, 256, 0, stream>>>(x, partial);
  msd_finalize_kernel<<<(TD_MAX_ + 255) / 256, 256, 0, stream>>>(partial, out);
}